// HierarchicalNode_72387378807011
// MI455X (gfx1250) — compile-verified
//
#include <hip/hip_runtime.h>
#include <hip/hip_bf16.h>
#include <math.h>

typedef float v2f __attribute__((ext_vector_type(2)));
typedef float v8f __attribute__((ext_vector_type(8)));

#define BQ 512   // batch
#define NN 128   // nodes
#define EE 512   // edges
#define DD 256   // feature dim
#define VV 128   // vocab

static constexpr float RSQ = 0.9999950000374997f; // 1/sqrt(1+1e-5)

// ---------------------------------------------------------------------------
// fp32 WMMA GEMM: out[M,256] = epi( (A1 [+ alpha*A2]) @ W + bias ), K=N=256.
// Block = 256 threads = 8 waves; block tile 64x64; wave tile 16x32 (two
// v_wmma_f32_16x16x4_f32 accumulators). Software pipeline: b128 global loads
// for chunk s+1 issue before the WMMA burst of chunk s (latency hidden by
// matrix math), double-buffered LDS, one barrier per 32-wide K chunk.
// Epilogue: ((acc+bias)*g*RSQ+beta), optional ReLU. FUSE: A = A1 + (1+eps)*A2.
// ---------------------------------------------------------------------------
template<bool FUSE>
__global__ __launch_bounds__(256) void gemm_bn_wmma(
    const float* __restrict__ A1, const float* __restrict__ A2,
    const float* __restrict__ epsp,
    const float* __restrict__ Wm,  const float* __restrict__ bias,
    const float* __restrict__ gvec, const float* __restrict__ bvec,
    float* __restrict__ out, int relu)
{
    __shared__ float As[2 * 64 * 34];   // [buf][row m][k], stride 34
    __shared__ float Bs[2 * 64 * 34];   // [buf][col n][k], transposed W tile

    const int tid  = threadIdx.x;
    const int gm0  = blockIdx.x * 64;
    const int n0   = blockIdx.y * 64;
    const int lane = tid & 31;
    const int w    = tid >> 5;
    const int wr16 = (w & 3) * 16;
    const int wc32 = (w >> 2) * 32;
    const int half = lane >> 4;         // K-pair selector per ISA A/B layout
    const int mr   = lane & 15;

    float alpha = 1.0f;
    if (FUSE) alpha = 1.0f + epsp[0];

    // per-thread fill coordinates (2 float4 each for A and W tiles)
    int offA[2], sA[2], offB[2], nB_[2], cB_[2];
    #pragma unroll
    for (int i = 0; i < 2; ++i) {
        int idx = tid + i * 256;
        int rA = idx >> 3, cA = (idx & 7) << 2;       // A: 64 rows x 32 k
        offA[i] = (gm0 + rA) * 256 + cA;
        sA[i]   = rA * 34 + cA;
        int nB = (idx & 15) << 2, cB = idx >> 4;      // W: 32 k x 64 n
        offB[i] = cB * 256 + n0 + nB;
        nB_[i] = nB; cB_[i] = cB;
    }

    v8f c0 = {};
    v8f c1 = {};
    float4 ra[2], ra2[2], rb[2];

    // prefetch + store chunk 0
    #pragma unroll
    for (int i = 0; i < 2; ++i) {
        ra[i] = *(const float4*)(A1 + offA[i]);
        if (FUSE) ra2[i] = *(const float4*)(A2 + offA[i]);
        rb[i] = *(const float4*)(Wm + offB[i]);
    }
    #pragma unroll
    for (int i = 0; i < 2; ++i) {
        float4 v = ra[i];
        if (FUSE) {
            v.x = fmaf(alpha, ra2[i].x, v.x); v.y = fmaf(alpha, ra2[i].y, v.y);
            v.z = fmaf(alpha, ra2[i].z, v.z); v.w = fmaf(alpha, ra2[i].w, v.w);
        }
        *(v2f*)&As[sA[i]]     = v2f{v.x, v.y};
        *(v2f*)&As[sA[i] + 2] = v2f{v.z, v.w};
        Bs[(nB_[i] + 0) * 34 + cB_[i]] = rb[i].x;
        Bs[(nB_[i] + 1) * 34 + cB_[i]] = rb[i].y;
        Bs[(nB_[i] + 2) * 34 + cB_[i]] = rb[i].z;
        Bs[(nB_[i] + 3) * 34 + cB_[i]] = rb[i].w;
    }
    __syncthreads();

    for (int s = 0; s < 8; ++s) {
        const int buf  = s & 1;
        const float* Asb = &As[buf * 2176];
        const float* Bsb = &Bs[buf * 2176];

        // issue global loads for chunk s+1 (in flight during WMMA burst)
        if (s < 7) {
            const int k1 = (s + 1) * 32;
            #pragma unroll
            for (int i = 0; i < 2; ++i) {
                ra[i] = *(const float4*)(A1 + offA[i] + k1);
                if (FUSE) ra2[i] = *(const float4*)(A2 + offA[i] + k1);
                rb[i] = *(const float4*)(Wm + offB[i] + k1 * 256);
            }
        }

        // WMMA burst over the current 32-wide K chunk
        #pragma unroll
        for (int kk = 0; kk < 32; kk += 4) {
            v2f a  = *(const v2f*)&Asb[(wr16 + mr) * 34 + kk + 2 * half];
            v2f b0 = *(const v2f*)&Bsb[(wc32 + mr) * 34 + kk + 2 * half];
            v2f b1 = *(const v2f*)&Bsb[(wc32 + 16 + mr) * 34 + kk + 2 * half];
            c0 = __builtin_amdgcn_wmma_f32_16x16x4_f32(false, a, false, b0,
                                                       (short)0, c0, false, false);
            c1 = __builtin_amdgcn_wmma_f32_16x16x4_f32(false, a, false, b1,
                                                       (short)0, c1, false, false);
        }

        // store chunk s+1 into the alternate LDS buffer
        if (s < 7) {
            float* Asn = &As[(buf ^ 1) * 2176];
            float* Bsn = &Bs[(buf ^ 1) * 2176];
            #pragma unroll
            for (int i = 0; i < 2; ++i) {
                float4 v = ra[i];
                if (FUSE) {
                    v.x = fmaf(alpha, ra2[i].x, v.x); v.y = fmaf(alpha, ra2[i].y, v.y);
                    v.z = fmaf(alpha, ra2[i].z, v.z); v.w = fmaf(alpha, ra2[i].w, v.w);
                }
                *(v2f*)&Asn[sA[i]]     = v2f{v.x, v.y};
                *(v2f*)&Asn[sA[i] + 2] = v2f{v.z, v.w};
                Bsn[(nB_[i] + 0) * 34 + cB_[i]] = rb[i].x;
                Bsn[(nB_[i] + 1) * 34 + cB_[i]] = rb[i].y;
                Bsn[(nB_[i] + 2) * 34 + cB_[i]] = rb[i].z;
                Bsn[(nB_[i] + 3) * 34 + cB_[i]] = rb[i].w;
            }
            __syncthreads();
        }
    }

    // C/D layout: VGPR v holds M = v + 8*(lane/16), N = lane%16
    const int col0 = n0 + wc32 + mr;
    const int col1 = col0 + 16;
    const float s0 = gvec[col0] * RSQ, sh0 = bvec[col0], bi0 = bias[col0];
    const float s1 = gvec[col1] * RSQ, sh1 = bvec[col1], bi1 = bias[col1];
    #pragma unroll
    for (int v = 0; v < 8; ++v) {
        int row = gm0 + wr16 + v + 8 * half;
        float x0 = (c0[v] + bi0) * s0 + sh0;
        float x1 = (c1[v] + bi1) * s1 + sh1;
        if (relu) { x0 = fmaxf(x0, 0.f); x1 = fmaxf(x1, 0.f); }
        out[row * 256 + col0] = x0;
        out[row * 256 + col1] = x1;
    }
}

// ---------------------------------------------------------------------------
// Elementwise / gather / scatter kernels
// ---------------------------------------------------------------------------
__global__ void atom_embed_k(const float* __restrict__ emb,
                             const int* __restrict__ x, float* __restrict__ h)
{
    int bn = blockIdx.x, d = threadIdx.x;
    const int* xf = x + bn * 9;
    float s = 0.f;
    #pragma unroll
    for (int f = 0; f < 9; ++f) s += emb[(f * VV + xf[f]) * DD + d];
    h[bn * DD + d] = s;
}

__global__ void vn_init_k(const float* __restrict__ vemb, float* __restrict__ vn)
{
    vn[blockIdx.x * DD + threadIdx.x] = vemb[threadIdx.x];
}

__global__ void vnadd_k(const float* __restrict__ h, const float* __restrict__ vn,
                        float* __restrict__ out, int Nc)
{
    int bn = blockIdx.x, d = threadIdx.x;
    int b = bn / Nc;
    out[bn * DD + d] = h[bn * DD + d] + vn[b * DD + d];
}

__global__ void zerof_k(float* __restrict__ p, int n)
{
    int i = blockIdx.x * blockDim.x + threadIdx.x;
    if (i < n) p[i] = 0.f;
}

__global__ void copyf_k(const float* __restrict__ a, float* __restrict__ b, int n)
{
    int i = blockIdx.x * blockDim.x + threadIdx.x;
    if (i < n) b[i] = a[i];
}

__global__ void edge_init_k(const int* __restrict__ ei, int* __restrict__ src,
                            int* __restrict__ tgt, float* __restrict__ em)
{
    int i = blockIdx.x * blockDim.x + threadIdx.x;
    if (i < BQ * EE) {
        src[i] = ei[2 * i];
        tgt[i] = ei[2 * i + 1];
        em[i]  = 1.0f;
    }
}

// msg = relu(h[src] + bond_emb(edge_attr)) scattered to aggr[tgt] (atomic f32)
__global__ void message_k(const float* __restrict__ h, const float* __restrict__ bemb,
                          const int* __restrict__ ea, const int* __restrict__ src,
                          const int* __restrict__ tgt, const float* __restrict__ em,
                          float* __restrict__ aggr, int Nc)
{
    int be = blockIdx.x;
    if (em[be] == 0.f) return;             // masked edges contribute 0
    int b = be >> 9;                       // E = 512
    int s = src[be], t = tgt[be];
    int d = threadIdx.x;
    const int* e3 = ea + be * 3;
    float bo = bemb[(0 * VV + e3[0]) * DD + d]
             + bemb[(1 * VV + e3[1]) * DD + d]
             + bemb[(2 * VV + e3[2]) * DD + d];
    float m = fmaxf(h[(b * Nc + s) * DD + d] + bo, 0.f);
    atomicAdd(&aggr[(b * Nc + t) * DD + d], m);
}

// v[b,d] = vn[b,d] + sum_n h_list[layer][b,n,d]
__global__ void vnsum_k(const float* __restrict__ hsl, const float* __restrict__ vn,
                        float* __restrict__ v, int Nc)
{
    int b = blockIdx.x, d = threadIdx.x;
    float s = vn[b * DD + d];
    for (int n = 0; n < Nc; ++n) s += hsl[(b * Nc + n) * DD + d];
    v[b * DD + d] = s;
}

// TopK pooling: score = tanh((h.w)/||w||); select K=Nc/2, build mapping.
__global__ __launch_bounds__(128) void topk_k(
    const float* __restrict__ h, const float* __restrict__ w,
    float* __restrict__ vals, int* __restrict__ idxK, int* __restrict__ map, int Nc)
{
    __shared__ float sc[128];
    __shared__ float sv[128];
    __shared__ int   si[128];
    int b = blockIdx.x, n = threadIdx.x;
    int K = Nc >> 1;

    float nrm = 0.f;
    for (int d = 0; d < DD; ++d) { float wd = w[d]; nrm += wd * wd; }
    nrm = sqrtf(nrm);

    float sco = -3.4e38f;
    if (n < Nc) {
        const float* hp = h + (b * Nc + n) * DD;
        float dot = 0.f;
        for (int d = 0; d < DD; ++d) dot += hp[d] * w[d];
        sco = tanhf(dot / nrm);
        map[b * Nc + n] = -1;
    }
    sc[n] = sco;
    __syncthreads();

    for (int j = 0; j < K; ++j) {
        sv[n] = sc[n]; si[n] = n;
        __syncthreads();
        for (int step = 64; step > 0; step >>= 1) {
            if (n < step) {
                float ov = sv[n + step]; int oi = si[n + step];
                if (ov > sv[n] || (ov == sv[n] && oi < si[n])) { sv[n] = ov; si[n] = oi; }
            }
            __syncthreads();
        }
        if (n == 0) {
            int best = si[0];
            vals[b * K + j] = sv[0];
            idxK[b * K + j] = best;
            map[b * Nc + best] = j;
            sc[best] = -3.4e38f;
        }
        __syncthreads();
    }
}

__global__ void gather_k(const float* __restrict__ hin, const float* __restrict__ vals,
                         const int* __restrict__ idxK, float* __restrict__ hout,
                         int Nc, int K)
{
    int bj = blockIdx.x, d = threadIdx.x;
    int b = bj / K, j = bj % K;
    int sn = idxK[b * K + j];
    float v = vals[b * K + j];
    hout[(b * K + j) * DD + d] = hin[(b * Nc + sn) * DD + d] * v;
}

__global__ void remap_k(int* __restrict__ src, int* __restrict__ tgt,
                        float* __restrict__ em, const int* __restrict__ map, int Nc)
{
    int i = blockIdx.x * blockDim.x + threadIdx.x;
    if (i >= BQ * EE) return;
    int b = i >> 9;
    int ns = map[b * Nc + src[i]];
    int nt = map[b * Nc + tgt[i]];
    float e = em[i];
    e *= (ns >= 0 ? 1.f : 0.f) * (nt >= 0 ? 1.f : 0.f);
    em[i]  = e;
    src[i] = ns < 0 ? 0 : ns;
    tgt[i] = nt < 0 ? 0 : nt;
}

// ---------------------------------------------------------------------------
// Orchestration.
// Assumed d_in order (params dict insertion-first, JAX sorted-key flatten):
//  0 atom_emb | 1..6 bn[l].{be,g} | 7 bond_emb |
//  8+7l conv[l].{W1,W2,b1,b2,be1,eps,g1} | 29 pool_w | 30 vn_emb |
//  31+8l vn_mlp[l].{W1,W2,b1,b2,be1,be2,g1,g2} | 47 x | 48 edge_index | 49 edge_attr
// ---------------------------------------------------------------------------
extern "C" void kernel_launch(void* const* d_in, const int* in_sizes, int n_in,
                              void* d_out, int out_size, void* d_ws, size_t ws_size,
                              hipStream_t stream)
{
    (void)in_sizes; (void)n_in; (void)out_size; (void)ws_size;

    const float* atom_emb = (const float*)d_in[0];
    const float* bond_emb = (const float*)d_in[7];
    const float* pool_w   = (const float*)d_in[29];
    const float* vn_emb   = (const float*)d_in[30];
    const int*   x        = (const int*)d_in[47];
    const int*   ei       = (const int*)d_in[48];
    const int*   ea       = (const int*)d_in[49];

    float* out = (float*)d_out;

    // workspace carve-out (256B aligned)
    char* p = (char*)d_ws;
    auto alloc = [&](size_t nbytes) {
        char* r = p; p += (nbytes + 255) & ~(size_t)255; return r;
    };
    const size_t HMAX = (size_t)BQ * NN * DD;   // 16.78M floats
    float* hA    = (float*)alloc(HMAX * 4);
    float* hB    = (float*)alloc(HMAX * 4);
    float* tmp   = (float*)alloc(HMAX * 4);
    float* aggr  = (float*)alloc(HMAX * 4);
    float* vn    = (float*)alloc((size_t)BQ * DD * 4);
    float* vbuf  = (float*)alloc((size_t)BQ * DD * 4);
    float* vbuf2 = (float*)alloc((size_t)BQ * DD * 4);
    float* vals  = (float*)alloc((size_t)BQ * 64 * 4);
    float* emask = (float*)alloc((size_t)BQ * EE * 4);
    int*   idxK  = (int*)alloc((size_t)BQ * 64 * 4);
    int*   map   = (int*)alloc((size_t)BQ * NN * 4);
    int*   srcb  = (int*)alloc((size_t)BQ * EE * 4);
    int*   tgtb  = (int*)alloc((size_t)BQ * EE * 4);

    // output slices double as h_list storage
    float* outslice[3] = { out,
                           out + (size_t)BQ * 128 * DD,
                           out + (size_t)BQ * (128 + 64) * DD };
    float* outfinal = out + (size_t)BQ * (128 + 64 + 32) * DD;

    // initial state
    atom_embed_k<<<BQ * NN, DD, 0, stream>>>(atom_emb, x, hA);
    vn_init_k<<<BQ, DD, 0, stream>>>(vn_emb, vn);
    edge_init_k<<<(BQ * EE + 255) / 256, 256, 0, stream>>>(ei, srcb, tgtb, emask);

    const int Ncurs[3] = {128, 64, 32};
    for (int layer = 0; layer < 3; ++layer) {
        const int Nc = Ncurs[layer];
        const int M  = BQ * Nc;
        float* hsl = outslice[layer];

        const int cb = 8 + 7 * layer;
        const float* W1  = (const float*)d_in[cb + 0];
        const float* W2  = (const float*)d_in[cb + 1];
        const float* b1  = (const float*)d_in[cb + 2];
        const float* b2  = (const float*)d_in[cb + 3];
        const float* be1 = (const float*)d_in[cb + 4];
        const float* eps = (const float*)d_in[cb + 5];
        const float* g1  = (const float*)d_in[cb + 6];
        const float* bnb = (const float*)d_in[1 + 2 * layer]; // bn[l].be
        const float* bng = (const float*)d_in[2 + 2 * layer]; // bn[l].g

        // h = h + vn  (stored to h_list[layer] slice, used onward)
        vnadd_k<<<M, DD, 0, stream>>>(hA, vn, hsl, Nc);

        // aggr = scatter_add(relu(h[src] + bond_emb))
        zerof_k<<<((size_t)M * DD + 255) / 256, 256, 0, stream>>>(aggr, M * DD);
        message_k<<<BQ * EE, DD, 0, stream>>>(hsl, bond_emb, ea, srcb, tgtb,
                                              emask, aggr, Nc);

        // tmp = relu(bn1((aggr + (1+eps)*h) @ W1 + b1))
        gemm_bn_wmma<true><<<dim3(M / 64, 4), 256, 0, stream>>>(
            aggr, hsl, eps, W1, b1, g1, be1, tmp, 1);
        // hB = bn(tmp @ W2 + b2)  (+ relu on non-final layers)
        gemm_bn_wmma<false><<<dim3(M / 64, 4), 256, 0, stream>>>(
            tmp, nullptr, nullptr, W2, b2, bng, bnb, hB, layer < 2 ? 1 : 0);

        if (layer < 2) {
            const int qb = 31 + 8 * layer;
            const float* qW1  = (const float*)d_in[qb + 0];
            const float* qW2  = (const float*)d_in[qb + 1];
            const float* qb1  = (const float*)d_in[qb + 2];
            const float* qb2  = (const float*)d_in[qb + 3];
            const float* qbe1 = (const float*)d_in[qb + 4];
            const float* qbe2 = (const float*)d_in[qb + 5];
            const float* qg1  = (const float*)d_in[qb + 6];
            const float* qg2  = (const float*)d_in[qb + 7];

            // virtual-node update (uses h_list[layer] + old vn)
            vnsum_k<<<BQ, DD, 0, stream>>>(hsl, vn, vbuf, Nc);
            gemm_bn_wmma<false><<<dim3(BQ / 64, 4), 256, 0, stream>>>(
                vbuf, nullptr, nullptr, qW1, qb1, qg1, qbe1, vbuf2, 1);
            gemm_bn_wmma<false><<<dim3(BQ / 64, 4), 256, 0, stream>>>(
                vbuf2, nullptr, nullptr, qW2, qb2, qg2, qbe2, vn, 1);

            // TopK pooling on hB (post-ReLU h), gather + edge remap
            const int K = Nc >> 1;
            topk_k<<<BQ, 128, 0, stream>>>(hB, pool_w, vals, idxK, map, Nc);
            gather_k<<<BQ * K, DD, 0, stream>>>(hB, vals, idxK, hA, Nc, K);
            remap_k<<<(BQ * EE + 255) / 256, 256, 0, stream>>>(srcb, tgtb, emask,
                                                               map, Nc);
        } else {
            copyf_k<<<((size_t)BQ * 32 * DD + 255) / 256, 256, 0, stream>>>(
                hB, outfinal, BQ * 32 * DD);
        }
    }
}